// HyenaCascade_70875550318634
// MI455X (gfx1250) — compile-verified
//
#include <hip/hip_runtime.h>
#include <cstddef>

typedef __attribute__((ext_vector_type(2))) float v2f;
typedef __attribute__((ext_vector_type(4))) float v4f;
typedef __attribute__((ext_vector_type(8))) float v8f;

constexpr int B_ = 2, L_ = 8192, HID_ = 1024, HD_ = 64;
constexpr int CHUNK = 16, NCHUNK = L_ / CHUNK;
constexpr int LTILE = 32;
constexpr int WPB = 4; // waves per block in the scan kernels

// ---- workspace layout (float offsets) ----
constexpr size_t T_OFF = 0, TB_OFF = 256, P_OFF = 512, PB_OFF = 768,
                 Q_OFF = 1024, QB_OFF = 1280, P16_OFF = 1536;
constexpr size_t X1V_OFF = 2048;
constexpr size_t NTOT = (size_t)B_ * HID_ * L_;          // 16,777,216 floats
constexpr size_t X2_OFF = X1V_OFF + NTOT;
constexpr size_t YACC_OFF = X2_OFF + NTOT;

__device__ __forceinline__ v8f wmma_f32_16x16x4(v2f a, v2f b, v8f c) {
  // 8-arg pattern: (neg_a, A, neg_b, B, c_mod, C, reuse_a, reuse_b)
  return __builtin_amdgcn_wmma_f32_16x16x4_f32(false, a, false, b, (short)0, c,
                                               false, false);
}

__device__ __forceinline__ float powi(float x, int e) {
  float r = 1.0f;
  for (int i = 0; i < e; ++i) r *= x;
  return r;
}

// ---------------------------------------------------------------------------
// Stage 0: build the six 16x16 scan matrices + p^16 from poles/residues.
// ---------------------------------------------------------------------------
__global__ void setup_mats_kernel(const float* __restrict__ log_poles,
                                  const float* __restrict__ residues,
                                  float* __restrict__ ws) {
  __shared__ float p[16], r[16];
  const int t = threadIdx.x;
  if (t < 16) { p[t] = expf(log_poles[t]); r[t] = residues[t]; }
  __syncthreads();
  const int i = t >> 4, j = t & 15;
  float Tij = 0.f, Tbij = 0.f;
  if (i >= j) { const int d = i - j; float s = 0.f;
    for (int q = 0; q < 16; ++q) s += r[q] * powi(p[q], d); Tij = s; }
  if (j >= i) { const int d = j - i; float s = 0.f;
    for (int q = 0; q < 16; ++q) s += r[q] * powi(p[q], d); Tbij = s; }
  ws[T_OFF  + t] = Tij;                       // causal Toeplitz
  ws[TB_OFF + t] = Tbij;                      // anti-causal Toeplitz
  ws[P_OFF  + t] = r[j] * powi(p[j], i + 1);  // fwd state->output
  ws[PB_OFF + t] = r[j] * powi(p[j], 16 - i); // bwd state->output
  ws[Q_OFF  + t] = powi(p[i], 15 - j);        // fwd input->state
  ws[QB_OFF + t] = powi(p[i], j);             // bwd input->state
  if (t < 16) ws[P16_OFF + t] = powi(p[t], 16);
}

// ---------------------------------------------------------------------------
// Stage 1: bidirectional depthwise 3-tap FIR + column split + x1*v gate.
//   z[l] = w0*(s[l-2]+s[l+2]) + w1*(s[l-1]+s[l+1]) + 2*w2*s[l]
// Writes x1v and x2 in (B, HID, L) layout for the scan kernels.
// ---------------------------------------------------------------------------
__global__ void __launch_bounds__(256)
shortconv_kernel(const float* __restrict__ u, const float* __restrict__ w,
                 float* __restrict__ ws) {
  float* __restrict__ x1v = ws + X1V_OFF;
  float* __restrict__ x2o = ws + X2_OFF;
  const int b  = blockIdx.z;
  const int cc = blockIdx.y * 256 + threadIdx.x;   // output channel [0,1024)
  const int hd = cc >> 6, jj = cc & 63;
  const int c2 = hd * 3 * HD_ + jj;   // x2 source channel
  const int c1 = c2 + HD_;            // x1 source channel
  const int cv = c2 + 2 * HD_;        // v  source channel
  const float e1_0 = w[c1*3+0], e1_1 = w[c1*3+1], e1_2 = 2.f*w[c1*3+2];
  const float ev_0 = w[cv*3+0], ev_1 = w[cv*3+1], ev_2 = 2.f*w[cv*3+2];
  const float e2_0 = w[c2*3+0], e2_1 = w[c2*3+1], e2_2 = 2.f*w[c2*3+2];
  const int l0 = blockIdx.x * LTILE;
  float s1[5], sv[5], s2[5];
  #pragma unroll
  for (int q = 0; q < 4; ++q) {
    const int l = l0 - 2 + q;
    if (l >= 0) {
      const size_t base = ((size_t)b * L_ + l) * (3 * HID_);
      s1[q] = u[base + c1]; sv[q] = u[base + cv]; s2[q] = u[base + c2];
    } else { s1[q] = 0.f; sv[q] = 0.f; s2[q] = 0.f; }
  }
  const size_t orow = ((size_t)b * HID_ + cc) * L_;
  for (int i2 = 0; i2 < LTILE; ++i2) {
    const int l = l0 + i2, lp = l + 2;
    if (lp < L_) {
      const size_t base = ((size_t)b * L_ + lp) * (3 * HID_);
      s1[4] = u[base + c1]; sv[4] = u[base + cv]; s2[4] = u[base + c2];
    } else { s1[4] = 0.f; sv[4] = 0.f; s2[4] = 0.f; }
    const float z1 = e1_0*(s1[0]+s1[4]) + e1_1*(s1[1]+s1[3]) + e1_2*s1[2];
    const float zv = ev_0*(sv[0]+sv[4]) + ev_1*(sv[1]+sv[3]) + ev_2*sv[2];
    const float z2 = e2_0*(s2[0]+s2[4]) + e2_1*(s2[1]+s2[3]) + e2_2*s2[2];
    x1v[orow + l] = z1 * zv;
    x2o[orow + l] = z2;
    #pragma unroll
    for (int q = 0; q < 4; ++q) { s1[q]=s1[q+1]; sv[q]=sv[q+1]; s2[q]=s2[q+1]; }
  }
}

// ---------------------------------------------------------------------------
// Stage 2: forward chunked SSM scan via fp32 WMMA. One wave = 16-channel
// strip, sequential over 512 chunks. Y = T@X + P@S ; S' = p16*S + Q@X.
// Q@X arrives in C/D layout; bounce through LDS (transposed, stride-20 pad)
// to re-feed it as a B-operand next chunk.
// ---------------------------------------------------------------------------
__global__ void __launch_bounds__(WPB * 32)
scan_fwd_kernel(float* __restrict__ ws) {
  __shared__ float xch[WPB * 320];
  const int wave = threadIdx.x >> 5;
  const int lane = threadIdx.x & 31;
  const int n = lane & 15, hlf = lane >> 4;
  const int strip = blockIdx.x * WPB + wave;     // [0, 128)
  const int b  = strip >> 6;                     // HID/16 = 64 strips / batch
  const int cb = (strip & 63) << 4;
  v2f aT[4], aP[4], aQ[4], pp[4];
  #pragma unroll
  for (int k = 0; k < 4; ++k) {
    const int col = 4 * k + 2 * hlf;             // A-layout K pair
    aT[k] = *(const v2f*)(ws + T_OFF  + n * 16 + col);
    aP[k] = *(const v2f*)(ws + P_OFF  + n * 16 + col);
    aQ[k] = *(const v2f*)(ws + Q_OFF  + n * 16 + col);
    pp[k] = *(const v2f*)(ws + P16_OFF + col);
  }
  const float* __restrict__ x = ws + X1V_OFF + ((size_t)(b * HID_ + cb + n)) * L_;
  float* __restrict__ y       = ws + YACC_OFF + ((size_t)(b * HID_ + cb + n)) * L_;
  float* myl = xch + wave * 320 + n * 20;
  v2f sB[4] = {v2f{0.f,0.f}, v2f{0.f,0.f}, v2f{0.f,0.f}, v2f{0.f,0.f}};
  for (int ch = 0; ch < NCHUNK; ++ch) {
    const float* xc = x + ch * CHUNK;
    __builtin_prefetch(xc + 4 * CHUNK, 0, 1);
    v2f xb[4];
    #pragma unroll
    for (int k = 0; k < 4; ++k) xb[k] = *(const v2f*)(xc + 4 * k + 2 * hlf);
    v8f ya = {0.f,0.f,0.f,0.f,0.f,0.f,0.f,0.f};
    v8f ga = {0.f,0.f,0.f,0.f,0.f,0.f,0.f,0.f};
    #pragma unroll
    for (int k = 0; k < 4; ++k) ya = wmma_f32_16x16x4(aT[k], xb[k], ya);
    #pragma unroll
    for (int k = 0; k < 4; ++k) ya = wmma_f32_16x16x4(aP[k], sB[k], ya);
    #pragma unroll
    for (int k = 0; k < 4; ++k) ga = wmma_f32_16x16x4(aQ[k], xb[k], ga);
    // C/D -> B layout swap through LDS (same-wave, in-order DS pipeline)
    *(v4f*)(myl + hlf * 8)     = (v4f){ga[0], ga[1], ga[2], ga[3]};
    *(v4f*)(myl + hlf * 8 + 4) = (v4f){ga[4], ga[5], ga[6], ga[7]};
    #pragma unroll
    for (int k = 0; k < 4; ++k) {
      const v2f g = *(const v2f*)(myl + 4 * k + 2 * hlf);
      sB[k] = pp[k] * sB[k] + g;
    }
    float* yc = y + ch * CHUNK + hlf * 8;
    *(v4f*)(yc)     = (v4f){ya[0], ya[1], ya[2], ya[3]};
    *(v4f*)(yc + 4) = (v4f){ya[4], ya[5], ya[6], ya[7]};
  }
}

// ---------------------------------------------------------------------------
// Stage 3: backward scan (time-reversed), folds in y_fwd, skip term D,
// output gate x2, and the (B,L,HID) transpose on the final store.
// ---------------------------------------------------------------------------
__global__ void __launch_bounds__(WPB * 32)
scan_bwd_kernel(float* __restrict__ ws, const float* __restrict__ Dv,
                float* __restrict__ out) {
  __shared__ float xch[WPB * 320];
  const int wave = threadIdx.x >> 5;
  const int lane = threadIdx.x & 31;
  const int n = lane & 15, hlf = lane >> 4;
  const int strip = blockIdx.x * WPB + wave;
  const int b  = strip >> 6;
  const int cb = (strip & 63) << 4;
  v2f aT[4], aP[4], aQ[4], pp[4];
  #pragma unroll
  for (int k = 0; k < 4; ++k) {
    const int col = 4 * k + 2 * hlf;
    aT[k] = *(const v2f*)(ws + TB_OFF + n * 16 + col);
    aP[k] = *(const v2f*)(ws + PB_OFF + n * 16 + col);
    aQ[k] = *(const v2f*)(ws + QB_OFF + n * 16 + col);
    pp[k] = *(const v2f*)(ws + P16_OFF + col);
  }
  const size_t row = ((size_t)(b * HID_ + cb + n)) * L_;
  const float* __restrict__ x  = ws + X1V_OFF  + row;
  const float* __restrict__ x2 = ws + X2_OFF   + row;
  const float* __restrict__ yf = ws + YACC_OFF + row;
  const float Dc = Dv[cb + n];
  float* myl = xch + wave * 320 + n * 20;
  v2f sB[4] = {v2f{0.f,0.f}, v2f{0.f,0.f}, v2f{0.f,0.f}, v2f{0.f,0.f}};
  for (int ch = NCHUNK - 1; ch >= 0; --ch) {
    const float* xc = x + ch * CHUNK;
    __builtin_prefetch(xc - 4 * CHUNK, 0, 1);
    v2f xb[4];
    #pragma unroll
    for (int k = 0; k < 4; ++k) {               // time-reversed B-layout load
      const v2f tp = *(const v2f*)(xc + 14 - 4 * k - 2 * hlf);
      xb[k] = (v2f){tp[1], tp[0]};
    }
    v8f ya = {0.f,0.f,0.f,0.f,0.f,0.f,0.f,0.f};
    v8f ga = {0.f,0.f,0.f,0.f,0.f,0.f,0.f,0.f};
    #pragma unroll
    for (int k = 0; k < 4; ++k) ya = wmma_f32_16x16x4(aT[k], xb[k], ya);
    #pragma unroll
    for (int k = 0; k < 4; ++k) ya = wmma_f32_16x16x4(aP[k], sB[k], ya);
    #pragma unroll
    for (int k = 0; k < 4; ++k) ga = wmma_f32_16x16x4(aQ[k], xb[k], ga);
    *(v4f*)(myl + hlf * 8)     = (v4f){ga[0], ga[1], ga[2], ga[3]};
    *(v4f*)(myl + hlf * 8 + 4) = (v4f){ga[4], ga[5], ga[6], ga[7]};
    #pragma unroll
    for (int k = 0; k < 4; ++k) {
      const v2f g = *(const v2f*)(myl + 4 * k + 2 * hlf);
      sB[k] = pp[k] * sB[k] + g;
    }
    // combine: out = (y_fwd + y_bwd + D*x1v) * x2, transposed store
    const int rb = ch * CHUNK + 8 * (1 - hlf);  // real-time block for this half
    const v4f f0 = *(const v4f*)(yf + rb), f1 = *(const v4f*)(yf + rb + 4);
    const v4f a0 = *(const v4f*)(x  + rb), a1 = *(const v4f*)(x  + rb + 4);
    const v4f g0 = *(const v4f*)(x2 + rb), g1 = *(const v4f*)(x2 + rb + 4);
    const float yfa[8] = {f0[0],f0[1],f0[2],f0[3],f1[0],f1[1],f1[2],f1[3]};
    const float x1a[8] = {a0[0],a0[1],a0[2],a0[3],a1[0],a1[1],a1[2],a1[3]};
    const float x2a[8] = {g0[0],g0[1],g0[2],g0[3],g1[0],g1[1],g1[2],g1[3]};
    const size_t ob = (size_t)b * L_ * HID_ + (size_t)(cb + n);
    #pragma unroll
    for (int v = 0; v < 8; ++v) {
      const int treal = ch * CHUNK + 15 - 8 * hlf - v;   // = rb + 7 - v
      const float val = (ya[v] + yfa[7 - v] + Dc * x1a[7 - v]) * x2a[7 - v];
      out[ob + (size_t)treal * HID_] = val;
    }
  }
}

// ---------------------------------------------------------------------------
extern "C" void kernel_launch(void* const* d_in, const int* in_sizes, int n_in,
                              void* d_out, int out_size, void* d_ws,
                              size_t ws_size, hipStream_t stream) {
  const float* u   = (const float*)d_in[0];
  const float* sfw = (const float*)d_in[1];
  const float* lp  = (const float*)d_in[2];
  const float* rs  = (const float*)d_in[3];
  const float* Dv  = (const float*)d_in[4];
  float* out = (float*)d_out;
  float* ws  = (float*)d_ws;
  (void)in_sizes; (void)n_in; (void)out_size; (void)ws_size;

  setup_mats_kernel<<<1, 256, 0, stream>>>(lp, rs, ws);
  shortconv_kernel<<<dim3(L_ / LTILE, HID_ / 256, B_), 256, 0, stream>>>(u, sfw, ws);
  scan_fwd_kernel<<<(B_ * (HID_ / 16)) / WPB, WPB * 32, 0, stream>>>(ws);
  scan_bwd_kernel<<<(B_ * (HID_ / 16)) / WPB, WPB * 32, 0, stream>>>(ws, Dv, out);
}